// DeepInsightEncoding_64493228916806
// MI455X (gfx1250) — compile-verified
//
#include <hip/hip_runtime.h>
#include <hip/hip_bf16.h>

// Problem constants (match reference)
#define B_  512
#define D_  32
#define H_  128
#define W_  128
#define C_  5
#define TR  8                       // rows per block tile
#define ROWF (W_ * C_)              // 640 floats per output row

static_assert(H_ % D_ == 0 && W_ % D_ == 0, "repeat factors assumed uniform");
static_assert(ROWF % 4 == 0, "row span must be float4-aligned");

// Equidistant bar layout (static, from reference _static_indices)
constexpr int BAR_W   = W_ / (3 * D_ + 2);            // 1
constexpr int TOTGAP  = W_ - D_ * BAR_W;              // 96
constexpr int GAP     = TOTGAP / (D_ + 1);            // 2
constexpr int EXTRA   = TOTGAP - GAP * (D_ + 1);      // 30
constexpr int BEG     = EXTRA / 2;                    // 15
constexpr int BAR_P   = BAR_W + GAP;                  // 3
constexpr int BAR_OFF = BEG + GAP;                    // 17: first bar column

// ---- CDNA5 async global->LDS path (ASYNCcnt), guarded so compile never breaks
#if __has_builtin(__builtin_amdgcn_global_load_async_to_lds_b128) && \
    __has_builtin(__builtin_amdgcn_global_load_async_to_lds_b32)
#define USE_ASYNC_LDS 1
#else
#define USE_ASYNC_LDS 0
#endif

#if __has_builtin(__builtin_amdgcn_s_wait_asynccnt)
#define WAIT_ASYNC() __builtin_amdgcn_s_wait_asynccnt(0)
#else
#define WAIT_ASYNC() asm volatile("s_wait_asynccnt 0" ::: "memory")
#endif

// Builtin-matching pointer types (probe-confirmed signatures:
//  b128: (vector_size(16) int * AS1, * AS3, Imm, Imm); b32: (int* AS1, int* AS3, Imm, Imm))
typedef int v4i __attribute__((vector_size(16)));
typedef __attribute__((address_space(1))) v4i* g_v4i;
typedef __attribute__((address_space(3))) v4i* l_v4i;
typedef __attribute__((address_space(1))) int* g_i32;
typedef __attribute__((address_space(3))) int* l_i32;

__global__ __launch_bounds__(256) void di_main(const float* __restrict__ inputs,
                                               const float* __restrict__ stamp,
                                               float* __restrict__ out) {
  __shared__ float s_x[D_];           // this batch's 32 inputs
  __shared__ float s_nd[2][D_];       // the two nd rows this tile touches
  __shared__ int   s_barh[D_];        // bar heights
  __shared__ float s_stamp[TR * W_];  // 8 stamp rows (4 KB)

  const int tid   = threadIdx.x;
  const int blk   = blockIdx.x;
  const int b     = blk >> 4;          // H_/TR = 16 tiles per batch
  const int htile = (blk & 15) * TR;

  // ---- Stage per-tile data into LDS (CDNA5 async global->LDS, ASYNCcnt) ----
#if USE_ASYNC_LDS
  {
    // 256 lanes x 16B = 4KB stamp tile (16B aligned: htile*W_*4 % 16 == 0)
    __builtin_amdgcn_global_load_async_to_lds_b128(
        (g_v4i)(stamp + (size_t)htile * W_) + tid,
        (l_v4i)s_stamp + tid, 0, 0);
    if (tid < D_) {
      __builtin_amdgcn_global_load_async_to_lds_b32(
          (g_i32)(inputs + (size_t)b * D_) + tid,
          (l_i32)s_x + tid, 0, 0);
    }
    WAIT_ASYNC();
  }
#else
  ((float4*)s_stamp)[tid] = ((const float4*)(stamp + (size_t)htile * W_))[tid];
  if (tid < D_) s_x[tid] = inputs[(size_t)b * D_ + tid];
#endif
  __syncthreads();

  // ---- Per-tile derived values ----
  if (tid < D_) {
    // bar_height = clip(round(x*H), 0, H); rintf == round-half-even (RNE)
    float v = rintf(s_x[tid] * (float)H_);
    v = fminf(fmaxf(v, 0.0f), (float)H_);
    s_barh[tid] = (int)v;
  }
  if (tid < 2 * D_) {
    // nd[i][j] = |x[row]-x[j]| / (max(x)-min(x))
    //   (min(dm)==0 exactly via diagonal; max(dm)==fl(max-min) exactly,
    //    since fp subtraction rounding is monotone)
    const int i    = tid >> 5;                  // 0..1
    const int j    = tid & (D_ - 1);
    const int rowd = (htile >> 2) + i;          // h>>2 takes 2 values in tile
    float mn = s_x[0], mx = s_x[0];
#pragma unroll
    for (int k = 1; k < D_; ++k) {
      const float v = s_x[k];
      mn = fminf(mn, v);
      mx = fmaxf(mx, v);
    }
    const float range = mx - mn;                // div-by-0 -> NaN, matches ref
    s_nd[i][j] = fabsf(s_x[rowd] - s_x[j]) / range;
  }
  __syncthreads();

  // ---- Streaming writes ----
  // Each thread owns 4 consecutive pixels = 20 floats = 5 float4s (80 B,
  // 16B-aligned). All div/mod and channel positions are compile-time:
  //   r  = tid>>5 (single row per thread), w = 4*(tid&31)+j,
  //   w>>2 == tid&31  -> one nd value per thread.
  const int r  = tid >> 5;                      // 0..7
  const int h  = htile + r;
  const int wb = (tid & 31) * 4;                // first of 4 columns
  const float rv = s_x[(htile >> 2) + (r >> 2)];
  const float nd = s_nd[r >> 2][tid & 31];

  // stamp for the 4 columns: one LDS b128 load
  const float4 st = ((const float4*)s_stamp)[(r * W_ + wb) >> 2];

  // bar value per column (membership static-per-thread arithmetic)
  float bar[4];
#pragma unroll
  for (int j = 0; j < 4; ++j) {
    const int w  = wb + j;
    const int t  = w - BAR_OFF;
    const int bi = t / BAR_P;                   // trunc toward 0
    const bool in = (t >= 0) && (t - bi * BAR_P < BAR_W) && (bi < D_);
    const int  ci = bi < 0 ? 0 : (bi >= D_ ? D_ - 1 : bi);
    bar[j] = (in && (h < s_barh[ci])) ? 1.0f : 0.0f;
  }

  // Pack 4 pixels x [stamp, 0, rowcopy, nd, bar] into 5 float4s (static layout)
  float4* out4 = (float4*)out;
  const size_t base4 = (size_t)(b * H_ + htile) * (ROWF / 4) + (size_t)tid * 5;
  out4[base4 + 0] = make_float4(st.x,   0.0f,  rv,    nd);
  out4[base4 + 1] = make_float4(bar[0], st.y,  0.0f,  rv);
  out4[base4 + 2] = make_float4(nd,     bar[1], st.z, 0.0f);
  out4[base4 + 3] = make_float4(rv,     nd,    bar[2], st.w);
  out4[base4 + 4] = make_float4(0.0f,   rv,    nd,    bar[3]);
}

// Deterministic duplicate-summed scatter into channel 1 (runs after di_main).
__global__ __launch_bounds__(256) void di_scatter(const float* __restrict__ inputs,
                                                  const int* __restrict__ coords,
                                                  float* __restrict__ out) {
  const int g = blockIdx.x * blockDim.x + threadIdx.x;
  if (g >= B_ * D_) return;
  const int b = g >> 5;
  const int d = g & (D_ - 1);
  const int r = coords[2 * d];
  const int c = coords[2 * d + 1];
  // only the first occurrence of (r,c) writes; it sums duplicates in index order
  for (int d2 = 0; d2 < d; ++d2)
    if (coords[2 * d2] == r && coords[2 * d2 + 1] == c) return;
  float val = 0.0f;
  for (int d2 = d; d2 < D_; ++d2)
    if (coords[2 * d2] == r && coords[2 * d2 + 1] == c)
      val += inputs[(size_t)b * D_ + d2];
  out[(((size_t)b * H_ + r) * W_ + c) * C_ + 1] = val;
}

extern "C" void kernel_launch(void* const* d_in, const int* in_sizes, int n_in,
                              void* d_out, int out_size, void* d_ws, size_t ws_size,
                              hipStream_t stream) {
  const float* inputs = (const float*)d_in[0];   // [B, D] f32
  const float* stamp  = (const float*)d_in[1];   // [H, W, 1] f32
  const int*   coords = (const int*)d_in[2];     // [D, 2] i32
  float* out = (float*)d_out;                    // [B, H, W, 5] f32

  di_main<<<B_ * (H_ / TR), 256, 0, stream>>>(inputs, stamp, out);
  di_scatter<<<(B_ * D_ + 255) / 256, 256, 0, stream>>>(inputs, coords, out);
}